// simple_Net_3650722202403
// MI455X (gfx1250) — compile-verified
//
#include <hip/hip_runtime.h>

#define B_TOTAL   32768
#define T_IN      1024
#define STEPS     512
#define BLOCK     256
#define CHUNK     32
#define LDS_PITCH (CHUNK + 1)   // 33: odd pitch -> conflict-free column reads

typedef float v2f __attribute__((ext_vector_type(2)));
typedef float v8f __attribute__((ext_vector_type(8)));

#if __has_builtin(__builtin_amdgcn_global_load_async_to_lds_b32)
#define HAVE_ASYNC_LDS 1
#else
#define HAVE_ASYNC_LDS 0
#endif

// ---- CDNA5 fast activations -------------------------------------------------
__device__ __forceinline__ float fast_tanh(float x) {
#if __has_builtin(__builtin_amdgcn_tanhf)
    return __builtin_amdgcn_tanhf(x);            // v_tanh_f32 (gfx1250)
#else
    // tanh(x) = 1 - 2/(2^(2*log2e*x) + 1)  via v_exp_f32 / v_rcp_f32
    float e = __builtin_amdgcn_exp2f(2.8853900817779268f * x);
    return 1.0f - 2.0f * __builtin_amdgcn_rcpf(e + 1.0f);
#endif
}

__device__ __forceinline__ float fast_sigmoid(float x) {
    return fmaf(0.5f, fast_tanh(0.5f * x), 0.5f);
}

// ---- LSTM cell, general input (open loop + first closed step) ---------------
__device__ __forceinline__ void lstm_cell(
    float xv, float& h, float& c,
    float wi0, float wi1, float wi2, float wi3,
    float wh0, float wh1, float wh2, float wh3,
    float bb0, float bb1, float bb2, float bb3) {
    float z0 = fmaf(wh0, h, fmaf(wi0, xv, bb0));
    float z1 = fmaf(wh1, h, fmaf(wi1, xv, bb1));
    float z2 = fmaf(wh2, h, fmaf(wi2, xv, bb2));
    float z3 = fmaf(wh3, h, fmaf(wi3, xv, bb3));
    float ig = fast_sigmoid(z0);
    float fg = fast_sigmoid(z1);
    float gg = fast_tanh(z2);
    float og = fast_sigmoid(z3);
    c = fmaf(fg, c, ig * gg);
    h = og * fast_tanh(c);
}

// ---- LSTM cell, autoregressive (x_in == h): 1 FMA on the critical chain -----
__device__ __forceinline__ void lstm_cell_ar(
    float& h, float& c,
    float wc0, float wc1, float wc2, float wc3,   // wc_g = wi_g + wh_g
    float bb0, float bb1, float bb2, float bb3) {
    float z0 = fmaf(wc0, h, bb0);
    float z1 = fmaf(wc1, h, bb1);
    float z2 = fmaf(wc2, h, bb2);
    float z3 = fmaf(wc3, h, bb3);
    float ig = fast_sigmoid(z0);
    float fg = fast_sigmoid(z1);
    float gg = fast_tanh(z2);
    float og = fast_sigmoid(z3);
    c = fmaf(fg, c, ig * gg);
    h = og * fast_tanh(c);
}

// ---- main kernel ------------------------------------------------------------
__global__ __launch_bounds__(BLOCK) void lstm_autoreg_kernel(
    const float* __restrict__ x,    // [B, 1, T_IN]
    const float* __restrict__ t,    // [B, 1, STEPS]
    const float* __restrict__ h0,   // [B, 1]
    const float* __restrict__ c0,   // [B, 1]
    const float* __restrict__ w_ih, // [4]
    const float* __restrict__ w_hh, // [4]
    const float* __restrict__ b_ih, // [4]
    const float* __restrict__ b_hh, // [4]
    float* __restrict__ out,                 // [1 + B*STEPS]
    float* __restrict__ block_partials) {    // [gridDim.x]
    __shared__ float tile[BLOCK * LDS_PITCH];
    __shared__ float wave_sums[BLOCK / 32];

    const int tid   = threadIdx.x;
    const int bbase = blockIdx.x * BLOCK;
    const int b     = bbase + tid;

    // Uniform weights -> scalar loads
    const float wi0 = w_ih[0], wi1 = w_ih[1], wi2 = w_ih[2], wi3 = w_ih[3];
    const float wh0 = w_hh[0], wh1 = w_hh[1], wh2 = w_hh[2], wh3 = w_hh[3];
    const float bb0 = b_ih[0] + b_hh[0];
    const float bb1 = b_ih[1] + b_hh[1];
    const float bb2 = b_ih[2] + b_hh[2];
    const float bb3 = b_ih[3] + b_hh[3];
    const float wc0 = wi0 + wh0, wc1 = wi1 + wh1;
    const float wc2 = wi2 + wh2, wc3 = wi3 + wh3;

    float h = h0[b];
    float c = c0[b];
    float p = 0.0f;  // squared-error partial

    const int lrow = tid >> 5;   // 0..7   (8 rows per coalesced pass)
    const int lcol = tid & 31;   // 0..31  (consecutive lanes -> consecutive t)

    // ---------------- Phase 1: open loop over x -----------------------------
    for (int tc = 0; tc < T_IN; tc += CHUNK) {
        __syncthreads();  // previous tile fully consumed
        // Stage [256 rows x 32 t] tile, coalesced (128B contiguous per wave op)
        for (int i = 0; i < BLOCK / 8; ++i) {
            const int row = i * 8 + lrow;
            const size_t gi = (size_t)(bbase + row) * T_IN + (size_t)(tc + lcol);
            const int    li = row * LDS_PITCH + lcol;
#if HAVE_ASYNC_LDS
            __builtin_amdgcn_global_load_async_to_lds_b32(
                (__attribute__((address_space(1))) int*)(x + gi),
                (__attribute__((address_space(3))) int*)(tile + li),
                0, 0);
#else
            tile[li] = x[gi];
#endif
        }
#if HAVE_ASYNC_LDS
        asm volatile("s_wait_asynccnt 0" ::: "memory");
#endif
        __syncthreads();
#pragma unroll 8
        for (int j = 0; j < CHUNK; ++j) {
            const float xv = tile[tid * LDS_PITCH + j];
            lstm_cell(xv, h, c, wi0, wi1, wi2, wi3, wh0, wh1, wh2, wh3,
                      bb0, bb1, bb2, bb3);
        }
    }

    // first closed-loop input = x[:, :, -1] (still resident in the tile)
    const float x_last = tile[tid * LDS_PITCH + (CHUNK - 1)];

    // ---------------- Phase 2: autoregressive closed loop -------------------
    // Peeled first chunk: step 0 consumes x_last; afterwards x_in == h, so the
    // gate pre-activation is (wi+wh)*h + b -> one FMA on the serial chain.
    {
        __syncthreads();
        lstm_cell(x_last, h, c, wi0, wi1, wi2, wi3, wh0, wh1, wh2, wh3,
                  bb0, bb1, bb2, bb3);
        tile[tid * LDS_PITCH + 0] = h;
#pragma unroll 8
        for (int j = 1; j < CHUNK; ++j) {
            lstm_cell_ar(h, c, wc0, wc1, wc2, wc3, bb0, bb1, bb2, bb3);
            tile[tid * LDS_PITCH + j] = h;
        }
        __syncthreads();
        for (int i = 0; i < BLOCK / 8; ++i) {
            const int row = i * 8 + lrow;
            const size_t gi = (size_t)(bbase + row) * STEPS + (size_t)lcol;
            const float pred = tile[row * LDS_PITCH + lcol];
            const float e = pred - t[gi];
            p = fmaf(e, e, p);
            out[1 + gi] = pred;
        }
    }
    for (int sc = CHUNK; sc < STEPS; sc += CHUNK) {
        __syncthreads();  // previous coalesced pass done reading tile
#pragma unroll 8
        for (int j = 0; j < CHUNK; ++j) {
            lstm_cell_ar(h, c, wc0, wc1, wc2, wc3, bb0, bb1, bb2, bb3);
            tile[tid * LDS_PITCH + j] = h;  // own row only
        }
        __syncthreads();
        // Coalesced: read targets, write preds, accumulate squared error
        for (int i = 0; i < BLOCK / 8; ++i) {
            const int row = i * 8 + lrow;
            const size_t gi = (size_t)(bbase + row) * STEPS + (size_t)(sc + lcol);
            const float pred = tile[row * LDS_PITCH + lcol];
            const float e = pred - t[gi];
            p = fmaf(e, e, p);
            out[1 + gi] = pred;
        }
    }

    // ---------------- Deterministic loss reduction --------------------------
    p *= (1.0f / (float)B_TOTAL);   // mean over batch; summed over steps

    // Wave32 reduction via V_WMMA_F32_16X16X4_F32 against all-ones B:
    // A[m][0]=p(lane m), A[m][2]=p(lane m+16) -> D[m][n] = p_m + p_{m+16}
    v2f a;  a.x = p;   a.y = 0.0f;
    v2f bm; bm.x = 1.0f; bm.y = 1.0f;
    v8f acc = {};
    acc = __builtin_amdgcn_wmma_f32_16x16x4_f32(
        /*neg_a=*/false, a, /*neg_b=*/false, bm,
        /*c_mod=*/(short)0, acc, /*reuse_a=*/false, /*reuse_b=*/false);
    float s = acc[0] + acc[1] + acc[2] + acc[3] +
              acc[4] + acc[5] + acc[6] + acc[7];       // rows 0-7 (or 8-15)
    s += __shfl_xor(s, 16, 32);                        // fold both halves

    const int lane = tid & 31;
    const int wave = tid >> 5;
    if (lane == 0) wave_sums[wave] = s;
    __syncthreads();
    if (tid == 0) {
        float bs = 0.0f;
        for (int w = 0; w < BLOCK / 32; ++w) bs += wave_sums[w];  // fixed order
        block_partials[blockIdx.x] = bs;
    }
}

// ---- final fixed-order reduction -------------------------------------------
__global__ void loss_finalize_kernel(const float* __restrict__ partials,
                                     float* __restrict__ out, int n) {
    if (blockIdx.x == 0 && threadIdx.x == 0) {
        float s = 0.0f;
        for (int i = 0; i < n; ++i) s += partials[i];  // deterministic order
        out[0] = s;
    }
}

extern "C" void kernel_launch(void* const* d_in, const int* in_sizes, int n_in,
                              void* d_out, int out_size, void* d_ws, size_t ws_size,
                              hipStream_t stream) {
    const float* x    = (const float*)d_in[0];
    const float* t    = (const float*)d_in[1];
    const float* h0   = (const float*)d_in[2];
    const float* c0   = (const float*)d_in[3];
    const float* w_ih = (const float*)d_in[4];
    const float* w_hh = (const float*)d_in[5];
    const float* b_ih = (const float*)d_in[6];
    const float* b_hh = (const float*)d_in[7];
    float* out      = (float*)d_out;
    float* partials = (float*)d_ws;   // gridDim.x floats of scratch

    const int grid = B_TOTAL / BLOCK; // 128 blocks, 8 wave32s each
    lstm_autoreg_kernel<<<grid, BLOCK, 0, stream>>>(
        x, t, h0, c0, w_ih, w_hh, b_ih, b_hh, out, partials);
    loss_finalize_kernel<<<1, 32, 0, stream>>>(partials, out, grid);
}